// S5Cascade_80152679678391
// MI455X (gfx1250) — compile-verified
//
#include <hip/hip_runtime.h>
#include <hip/hip_bf16.h>
#include <math.h>

// ---------------------------------------------------------------------------
// S5 cascade for MI455X (gfx1250).
// Diagonal-SSM scan over L=16384 recast as chunk-16 convolution:
//   y[16c+i] = sum_{j<=i} K[i][j] u[16c+j]           (16x16 WMMA GEMM, f32)
//           + sum_p 2Re(C_p lam_p^{i+1} X_{c-1,p})   (carry readout)
//   X_c = lam^16 X_{c-1} + sum_j lam^{15-j} Bbar u[16c+j]   (tiny scan)
// K, W, E, lam^16 precomputed on-device per block from the S5 params.
// LayerNorm over H=1 computed exactly as written (collapses to bias).
// Tile staging uses GLOBAL_LOAD_ASYNC_TO_LDS_B128 + S_WAIT_ASYNCCNT.
// ---------------------------------------------------------------------------

#define BATCHN 32
#define SEQLEN 16384
#define NCHK   (SEQLEN / 16)   // 1024 chunks per batch row
#define MAXP   18
#define LN_EPS 1e-5f

// per-block parameter scratch layout (floats)
#define PRM_K     0      // 16x16 impulse-response kernel (row major)
#define PRM_WRE   256    // W[p][j] re, stride 16
#define PRM_WIM   544
#define PRM_ERE   832    // E[i][p] re, stride MAXP
#define PRM_EIM   1120
#define PRM_L16RE 1408   // lam_bar^16 per state
#define PRM_L16IM 1426
#define PRM_SIZE  1536

typedef __attribute__((ext_vector_type(2))) float v2f;
typedef __attribute__((ext_vector_type(8))) float v8f;

__device__ __forceinline__ float gelu_exact(float x) {
  return 0.5f * x * (1.0f + erff(x * 0.7071067811865475f));
}

// exact H=1 layernorm: mean = v, var = 0  ->  bias (computed as written)
__device__ __forceinline__ float ln_h1(float v, float g, float b) {
  return ((v - v) * rsqrtf(LN_EPS)) * g + b;
}

// ---------------------------------------------------------------------------
// Precompute per-block derived matrices from S5 parameters. One workgroup.
// ---------------------------------------------------------------------------
__global__ void s5_precompute(const float* __restrict__ Lre, const float* __restrict__ Lim,
                              const float* __restrict__ Bre, const float* __restrict__ Bim,
                              const float* __restrict__ Cre, const float* __restrict__ Cim,
                              const float* __restrict__ lstep,
                              float* __restrict__ prm, int P) {
  __shared__ float lpr[MAXP][17], lpi[MAXP][17];   // lam_bar^d, d=0..16
  __shared__ float bbr[MAXP], bbi[MAXP], ccr[MAXP], cci[MAXP];
  int t = threadIdx.x;
  if (t < P) {
    float st  = expf(lstep[t]);
    float lr  = Lre[t], li = Lim[t];
    float mag = expf(lr * st);
    float ang = li * st;
    float lbr = mag * cosf(ang), lbi = mag * sinf(ang);   // lam_bar
    float pr = 1.f, pi = 0.f;
    for (int d = 0; d <= 16; ++d) {
      lpr[t][d] = pr; lpi[t][d] = pi;
      float nr = pr * lbr - pi * lbi;
      float ni = pr * lbi + pi * lbr;
      pr = nr; pi = ni;
    }
    // Bbar = ((lam_bar - 1) / lam) * B
    float d2 = lr * lr + li * li;
    float qr = ((lbr - 1.f) * lr + lbi * li) / d2;
    float qi = (lbi * lr - (lbr - 1.f) * li) / d2;
    bbr[t] = qr * Bre[t] - qi * Bim[t];
    bbi[t] = qr * Bim[t] + qi * Bre[t];
    ccr[t] = Cre[t]; cci[t] = Cim[t];
    prm[PRM_L16RE + t] = lpr[t][16];
    prm[PRM_L16IM + t] = lpi[t][16];
  }
  __syncthreads();
  // K[i][j] = sum_p 2*Re(C_p * lam^{i-j} * Bbar_p) for i>=j
  for (int e = t; e < 256; e += blockDim.x) {
    int i = e >> 4, j = e & 15;
    float val = 0.f;
    if (i >= j) {
      int d = i - j;
      for (int p = 0; p < P; ++p) {
        float mr = lpr[p][d] * bbr[p] - lpi[p][d] * bbi[p];
        float mi = lpr[p][d] * bbi[p] + lpi[p][d] * bbr[p];
        val += 2.f * (ccr[p] * mr - cci[p] * mi);
      }
    }
    prm[PRM_K + e] = val;
  }
  // W[p][j] = Bbar_p * lam^{15-j}
  for (int e = t; e < P * 16; e += blockDim.x) {
    int p = e >> 4, j = e & 15, d = 15 - j;
    prm[PRM_WRE + p * 16 + j] = bbr[p] * lpr[p][d] - bbi[p] * lpi[p][d];
    prm[PRM_WIM + p * 16 + j] = bbr[p] * lpi[p][d] + bbi[p] * lpr[p][d];
  }
  // E[i][p] = 2 * C_p * lam^{i+1}
  for (int e = t; e < 16 * P; e += blockDim.x) {
    int i = e % 16, p = e / 16;
    prm[PRM_ERE + i * MAXP + p] = 2.f * (ccr[p] * lpr[p][i + 1] - cci[p] * lpi[p][i + 1]);
    prm[PRM_EIM + i * MAXP + p] = 2.f * (ccr[p] * lpi[p][i + 1] + cci[p] * lpr[p][i + 1]);
  }
}

// ---------------------------------------------------------------------------
// Bulk kernel: one wave per (batch row, 16-chunk tile).
// Tile staged into LDS with GLOBAL_LOAD_ASYNC_TO_LDS_B128 (ASYNCcnt-tracked),
// then Y_tile(16x16) = K(16x16) @ U_tile(16x16) via 4x v_wmma_f32_16x16x4_f32.
// Also emits per-chunk complex carry inputs S[b][c][p] = sum_j W[p][j] u[j].
// LDS tile is chunk-major Ucm[c][j], row stride 20 floats (80 B):
//   - each lane's B128 lands 16-byte aligned (80*c + 16*q)
//   - B-operand pair reads are 8-byte aligned
//   - 20-float stride is bank-conflict-free across 16 lanes (gcd(20,64)=4)
// ---------------------------------------------------------------------------
__global__ void __launch_bounds__(32)
s5_bulk(const float* __restrict__ vin,
        const float* __restrict__ ln1g, const float* __restrict__ ln1b,
        const float* __restrict__ prm,
        float* __restrict__ yp, float* __restrict__ S, int P) {
  __shared__ float Ucm[16][20];      // [chunk][pos], raw (pre-LN) values
  const int lane = threadIdx.x;
  const int b = blockIdx.x;          // 0..31
  const int chunk0 = blockIdx.y * 16;
  const float g = ln1g[0], bb = ln1b[0];
  const float* src = vin + (size_t)b * SEQLEN + (size_t)chunk0 * 16;

  // async copy: 64 groups of 4 floats, 2 per lane, memory->LDS directly
  unsigned lbase = (unsigned)(unsigned long long)(void*)&Ucm[0][0];
#pragma unroll
  for (int q = 0; q < 2; ++q) {
    int gidx = lane + 32 * q;        // 0..63
    int c = gidx >> 2, w = gidx & 3; // chunk, 4-float group within chunk
    unsigned loff = lbase + (unsigned)(c * 20 + 4 * w) * 4u;
    const float* gp = src + c * 16 + 4 * w;
    asm volatile("global_load_async_to_lds_b128 %0, %1, off"
                 :: "v"(loff), "v"(gp) : "memory");
  }
  asm volatile("s_wait_asynccnt 0x0" ::: "memory");
  __syncthreads();

  const int row = lane & 15;
  const int hi  = lane >> 4;
  v8f acc = {};
#pragma unroll
  for (int s = 0; s < 4; ++s) {
    v2f a, bm;
    // A (16x4 slice of K): lanes 0-15 hold K-cols {4s,4s+1}, lanes 16-31 {4s+2,4s+3}
    a.x = prm[PRM_K + row * 16 + 4 * s + 2 * hi];
    a.y = prm[PRM_K + row * 16 + 4 * s + 2 * hi + 1];
    // B (4x16 slice of U^T): B[k][n] = LN(Ucm[n][k]); VGPR0 rows {4s,4s+2}, VGPR1 {4s+1,4s+3}
    bm.x = ln_h1(Ucm[row][4 * s + 2 * hi], g, bb);
    bm.y = ln_h1(Ucm[row][4 * s + 2 * hi + 1], g, bb);
    acc = __builtin_amdgcn_wmma_f32_16x16x4_f32(
        /*neg_a=*/false, a, /*neg_b=*/false, bm,
        /*c_mod=*/(short)0, acc, /*reuse_a=*/false, /*reuse_b=*/false);
  }
  // D-matrix layout: VGPR r -> (M=r, N=lane) lanes 0-15; (M=r+8, N=lane-16) lanes 16-31
  float* dst = yp + (size_t)b * SEQLEN;
#pragma unroll
  for (int r = 0; r < 8; ++r) {
    int m = r + 8 * hi;
    dst[(chunk0 + row) * 16 + m] = acc[r];
  }
  // per-chunk carry inputs (lanes 0-15, one chunk each)
  if (hi == 0) {
    int c = row;
    for (int p = 0; p < P; ++p) {
      float sr = 0.f, si = 0.f;
#pragma unroll
      for (int j = 0; j < 16; ++j) {
        float u = ln_h1(Ucm[c][j], g, bb);
        sr += prm[PRM_WRE + p * 16 + j] * u;
        si += prm[PRM_WIM + p * 16 + j] * u;
      }
      size_t idx = ((size_t)(b * NCHK + chunk0 + c) * MAXP + p) * 2;
      S[idx] = sr; S[idx + 1] = si;
    }
  }
}

// ---------------------------------------------------------------------------
// Cross-chunk carry scan: 32*P independent complex recurrences of length 1024.
// car[b][c][p] = state entering chunk c.
// ---------------------------------------------------------------------------
__global__ void s5_scan(const float* __restrict__ prm, const float* __restrict__ S,
                        float* __restrict__ car, int P) {
  int t = blockIdx.x * blockDim.x + threadIdx.x;
  if (t >= BATCHN * P) return;
  int b = t / P, p = t % P;
  float ar = prm[PRM_L16RE + p], ai = prm[PRM_L16IM + p];
  float xr = 0.f, xi = 0.f;
  for (int c = 0; c < NCHK; ++c) {
    size_t idx = ((size_t)(b * NCHK + c) * MAXP + p) * 2;
    car[idx] = xr; car[idx + 1] = xi;
    float sr = S[idx], si = S[idx + 1];
    float nr = ar * xr - ai * xi + sr;
    float ni = ar * xi + ai * xr + si;
    xr = nr; xi = ni;
  }
}

// ---------------------------------------------------------------------------
// Finalize: y = Y_within + E@carry + D*fx;  x2 = gelu(y)+fx;  LN2 -> GEGLU FF.
// ---------------------------------------------------------------------------
__global__ void s5_finalize(const float* __restrict__ vin,
                            const float* __restrict__ yp,
                            const float* __restrict__ car,
                            const float* __restrict__ prm,
                            const float* __restrict__ Dp,
                            const float* __restrict__ ln1g, const float* __restrict__ ln1b,
                            const float* __restrict__ ln2g, const float* __restrict__ ln2b,
                            const float* __restrict__ Wenc, const float* __restrict__ Wdec,
                            float* __restrict__ vout, int P) {
  int tid = blockIdx.x * blockDim.x + threadIdx.x;
  if (tid >= BATCHN * SEQLEN) return;
  int b = tid / SEQLEN, l = tid % SEQLEN;
  int c = l >> 4, i = l & 15;
  float fx = ln_h1(vin[tid], ln1g[0], ln1b[0]);
  float y = yp[tid] + Dp[0] * fx;
  size_t cb = ((size_t)(b * NCHK + c) * MAXP) * 2;
  for (int p = 0; p < P; ++p) {
    float er = prm[PRM_ERE + i * MAXP + p];
    float ei = prm[PRM_EIM + i * MAXP + p];
    float xr = car[cb + 2 * p], xi = car[cb + 2 * p + 1];
    y += er * xr - ei * xi;
  }
  float x2  = gelu_exact(y) + fx;
  float fx2 = ln_h1(x2, ln2g[0], ln2b[0]);
  float aa  = fx2 * Wenc[0];
  float gt  = fx2 * Wenc[1];
  vout[tid] = aa * gelu_exact(gt) * Wdec[0] + fx2;
}

// ---------------------------------------------------------------------------
// Channel mix: out_o = tanh(sum_i in_i * W[i][o] + b[o]), up to 3 in/out.
// ---------------------------------------------------------------------------
__global__ void mix_tanh(const float* __restrict__ in0, const float* __restrict__ in1,
                         const float* __restrict__ in2,
                         const float* __restrict__ Wp, const float* __restrict__ bp,
                         int nin, int nout,
                         float* __restrict__ out0, float* __restrict__ out1,
                         float* __restrict__ out2) {
  int tid = blockIdx.x * blockDim.x + threadIdx.x;
  if (tid >= BATCHN * SEQLEN) return;
  float x0 = in0[tid];
  float x1 = (nin > 1) ? in1[tid] : 0.f;
  float x2 = (nin > 2) ? in2[tid] : 0.f;
  float r[3];
  for (int o = 0; o < nout; ++o) {
    float s = bp[o] + x0 * Wp[0 * nout + o];
    if (nin > 1) s += x1 * Wp[1 * nout + o];
    if (nin > 2) s += x2 * Wp[2 * nout + o];
    r[o] = tanhf(s);
  }
  out0[tid] = r[0];
  if (nout > 1) out1[tid] = r[1];
  if (nout > 2) out2[tid] = r[2];
}

// ---------------------------------------------------------------------------
// Host-side orchestration.
// ---------------------------------------------------------------------------
static const int PBLK[9] = {18, 12, 12, 6, 6, 6, 12, 12, 18};

static void run_block(hipStream_t stream, void* const* d_in, int bi,
                      const float* vin, float* vout,
                      float* prm_all, float* yp, float* Sbuf, float* car) {
  const int P = PBLK[bi];
  const int base = 1 + 14 * bi;   // insertion-order flattening of block dict
  const float* Lre   = (const float*)d_in[base + 0];
  const float* Lim   = (const float*)d_in[base + 1];
  const float* Bre   = (const float*)d_in[base + 2];
  const float* Bim   = (const float*)d_in[base + 3];
  const float* Cre   = (const float*)d_in[base + 4];
  const float* Cim   = (const float*)d_in[base + 5];
  const float* Dp    = (const float*)d_in[base + 6];
  const float* lstep = (const float*)d_in[base + 7];
  const float* ln1g  = (const float*)d_in[base + 8];
  const float* ln1b  = (const float*)d_in[base + 9];
  const float* ln2g  = (const float*)d_in[base + 10];
  const float* ln2b  = (const float*)d_in[base + 11];
  const float* Wenc  = (const float*)d_in[base + 12];
  const float* Wdec  = (const float*)d_in[base + 13];
  float* prm = prm_all + (size_t)bi * PRM_SIZE;

  s5_precompute<<<1, 256, 0, stream>>>(Lre, Lim, Bre, Bim, Cre, Cim, lstep, prm, P);
  dim3 gb(BATCHN, NCHK / 16);
  s5_bulk<<<gb, 32, 0, stream>>>(vin, ln1g, ln1b, prm, yp, Sbuf, P);
  int nt = BATCHN * P;
  s5_scan<<<(nt + 63) / 64, 64, 0, stream>>>(prm, Sbuf, car, P);
  s5_finalize<<<(BATCHN * SEQLEN) / 256, 256, 0, stream>>>(
      vin, yp, car, prm, Dp, ln1g, ln1b, ln2g, ln2b, Wenc, Wdec, vout, P);
}

extern "C" void kernel_launch(void* const* d_in, const int* in_sizes, int n_in,
                              void* d_out, int out_size, void* d_ws, size_t ws_size,
                              hipStream_t stream) {
  (void)in_sizes; (void)n_in; (void)out_size; (void)ws_size;
  const size_t N = (size_t)BATCHN * SEQLEN;

  float* ws = (float*)d_ws;
  float* prm_all = ws;                                 // 9 * 1536 floats
  float* fld[6];
  size_t off = 9 * (size_t)PRM_SIZE;
  for (int k = 0; k < 6; ++k) { fld[k] = ws + off; off += N; }
  float* yp   = ws + off; off += N;
  float* Sbuf = ws + off; off += (size_t)BATCHN * NCHK * MAXP * 2;
  float* car  = ws + off;

  const float* x = (const float*)d_in[0];
  const int LB = 1 + 9 * 14;                            // 127: lins base
  const float* W1 = (const float*)d_in[LB + 0];
  const float* b1 = (const float*)d_in[LB + 1];
  const float* W2 = (const float*)d_in[LB + 2];
  const float* b2 = (const float*)d_in[LB + 3];
  const float* W3 = (const float*)d_in[LB + 4];
  const float* b3 = (const float*)d_in[LB + 5];
  const float* W4 = (const float*)d_in[LB + 6];
  const float* b4 = (const float*)d_in[LB + 7];

  const int MG = (int)(N / 256);

  // stage 0: block0 on raw x
  run_block(stream, d_in, 0, x, fld[0], prm_all, yp, Sbuf, car);
  // 1 -> 2
  mix_tanh<<<MG, 256, 0, stream>>>(fld[0], nullptr, nullptr, W1, b1, 1, 2,
                                   fld[1], fld[2], nullptr);
  run_block(stream, d_in, 1, fld[1], fld[3], prm_all, yp, Sbuf, car);
  run_block(stream, d_in, 2, fld[2], fld[4], prm_all, yp, Sbuf, car);
  // 2 -> 3
  mix_tanh<<<MG, 256, 0, stream>>>(fld[3], fld[4], nullptr, W2, b2, 2, 3,
                                   fld[0], fld[1], fld[2]);
  run_block(stream, d_in, 3, fld[0], fld[3], prm_all, yp, Sbuf, car);
  run_block(stream, d_in, 4, fld[1], fld[4], prm_all, yp, Sbuf, car);
  run_block(stream, d_in, 5, fld[2], fld[5], prm_all, yp, Sbuf, car);
  // 3 -> 2
  mix_tanh<<<MG, 256, 0, stream>>>(fld[3], fld[4], fld[5], W3, b3, 3, 2,
                                   fld[0], fld[1], nullptr);
  run_block(stream, d_in, 6, fld[0], fld[2], prm_all, yp, Sbuf, car);
  run_block(stream, d_in, 7, fld[1], fld[3], prm_all, yp, Sbuf, car);
  // 2 -> 1 (tanh fused), then final block writes d_out
  mix_tanh<<<MG, 256, 0, stream>>>(fld[2], fld[3], nullptr, W4, b4, 2, 1,
                                   fld[4], nullptr, nullptr);
  run_block(stream, d_in, 8, fld[4], (float*)d_out, prm_all, yp, Sbuf, car);
}